// FernBitWord_37512244364035
// MI455X (gfx1250) — compile-verified
//
#include <hip/hip_runtime.h>
#include <hip/hip_bf16.h>

#define TS_X 64
#define TS_Y 32
#define HALO_Y 7
#define LDS_W 80            // x coverage [gx0-8, gx0+71], 20 x 16B chunks per row
#define LDS_H 46            // y coverage [gy0-7, gy0+38]
#define LDS_X_OFF 8
#define CHUNKS_X (LDS_W / 4)
#define NN 16
#define DD 8
#define HH 512
#define WW 512
#define MK 4

__global__ __launch_bounds__(256) void fern_bits_kernel(
    const float* __restrict__ x,
    const float* __restrict__ dx1, const float* __restrict__ dy1,
    const float* __restrict__ dx2, const float* __restrict__ dy2,
    const float* __restrict__ th,  const float* __restrict__ amb,
    const int*   __restrict__ ch_idx,
    float* __restrict__ out)
{
    __shared__ __align__(16) float tile[LDS_H * LDS_W];

    const int tid = threadIdx.x;
    const int gx0 = blockIdx.x * TS_X;
    const int gy0 = blockIdx.y * TS_Y;
    const int z   = blockIdx.z;        // n*MK + mk  (matches (N, M*K, H, W) layout)
    const int n   = z >> 2;
    const int mk  = z & 3;

    // ---- per-block uniform parameters ----
    const int   ch  = ch_idx[mk];
    const float a1x = dx1[mk], a1y = dy1[mk];
    const float a2x = dx2[mk], a2y = dy2[mk];
    const float tt  = th[mk];
    const float pos = amb[mk], neg = amb[MK + mk];   // amb shape (1,2,4): [pos(4), neg(4)]
    const float inv = 1.0f / (pos - neg);

    const float fl1x = floorf(a1x), fl1y = floorf(a1y);
    const float fl2x = floorf(a2x), fl2y = floorf(a2y);
    const float f1x = fabsf(a1x - fl1x), f1y = fabsf(a1y - fl1y);
    const float f2x = fabsf(a2x - fl2x), f2y = fabsf(a2y - fl2y);
    const int i1x = (int)fl1x, i1y = (int)fl1y;
    const int i2x = (int)fl2x, i2y = (int)fl2y;
    const float w1_00 = (1.f - f1x) * (1.f - f1y), w1_10 = f1x * (1.f - f1y);
    const float w1_01 = (1.f - f1x) * f1y,         w1_11 = f1x * f1y;
    const float w2_00 = (1.f - f2x) * (1.f - f2y), w2_10 = f2x * (1.f - f2y);
    const float w2_01 = (1.f - f2x) * f2y,         w2_11 = f2x * f2y;

    const float* chan = x + (size_t)((n * DD + ch) * HH) * WW;

    // ---- phase 1: async bulk-stage the halo tile with clamped (always valid)
    //      addresses; 16B chunks; each chunk is fully in- or out-of-image ----
    for (int s = tid; s < LDS_H * CHUNKS_X; s += 256) {
        const int r  = s / CHUNKS_X;
        const int c  = s - r * CHUNKS_X;
        int gy = gy0 - HALO_Y + r;
        int gx = gx0 - LDS_X_OFF + c * 4;            // 4-aligned
        gy = gy < 0 ? 0 : (gy > HH - 1 ? HH - 1 : gy);
        gx = gx < 0 ? 0 : (gx > WW - 4 ? WW - 4 : gx);
        // low 32 bits of a generic LDS pointer == LDS byte offset (flat aperture rule)
        unsigned lds_addr = (unsigned)(unsigned long long)(const void*)
                                &tile[r * LDS_W + c * 4];
        unsigned long long ga = (unsigned long long)(const void*)
                                (chan + (size_t)gy * WW + gx);
        asm volatile("global_load_async_to_lds_b128 %0, %1, off"
                     :: "v"(lds_addr), "v"(ga) : "memory");
    }
    asm volatile("s_wait_asynccnt 0x0" ::: "memory");

    // ---- phase 2: zero-fix only the out-of-image chunks (boundary blocks only) ----
    if (gy0 < HALO_Y || gy0 + TS_Y + HALO_Y + 1 > HH ||
        gx0 < LDS_X_OFF || gx0 + TS_X + LDS_X_OFF > WW) {
        const float4 z4 = make_float4(0.f, 0.f, 0.f, 0.f);
        for (int s = tid; s < LDS_H * CHUNKS_X; s += 256) {
            const int r  = s / CHUNKS_X;
            const int c  = s - r * CHUNKS_X;
            const int gy = gy0 - HALO_Y + r;
            const int gx = gx0 - LDS_X_OFF + c * 4;
            if (gy < 0 || gy >= HH || gx < 0 || gx >= WW)
                *(float4*)&tile[r * LDS_W + c * 4] = z4;
        }
        asm volatile("s_wait_dscnt 0x0" ::: "memory");
    }
    __syncthreads();

    // ---- phase 3: compute 64x32 outputs, 2 adjacent pixels per thread per iter ----
    const int tx2 = (tid & 31) * 2;    // x pair base: 0,2,...,62
    const int tyb = tid >> 5;          // 8 row groups
    float* outp = out + (size_t)z * HH * WW;

    #pragma unroll
    for (int i = 0; i < 4; ++i) {
        const int ty = tyb + i * 8;

        // interp A: 3 taps per row, rows r1 and r1+1 (middle tap shared)
        const int r1 = ty + i1y + HALO_Y, c1 = tx2 + i1x + LDS_X_OFF;
        const float* t1 = &tile[r1 * LDS_W + c1];
        const float a00 = t1[0], a01 = t1[1], a02 = t1[2];
        const float a10 = t1[LDS_W], a11 = t1[LDS_W + 1], a12 = t1[LDS_W + 2];
        const float vA0 = w1_00 * a00 + w1_10 * a01 + w1_01 * a10 + w1_11 * a11;
        const float vA1 = w1_00 * a01 + w1_10 * a02 + w1_01 * a11 + w1_11 * a12;

        const int r2 = ty + i2y + HALO_Y, c2 = tx2 + i2x + LDS_X_OFF;
        const float* t2 = &tile[r2 * LDS_W + c2];
        const float b00 = t2[0], b01 = t2[1], b02 = t2[2];
        const float b10 = t2[LDS_W], b11 = t2[LDS_W + 1], b12 = t2[LDS_W + 2];
        const float vB0 = w2_00 * b00 + w2_10 * b01 + w2_01 * b10 + w2_11 * b11;
        const float vB1 = w2_00 * b01 + w2_10 * b02 + w2_01 * b11 + w2_11 * b12;

        float u0 = (vA0 - vB0 - tt - neg) * inv;
        float u1 = (vA1 - vB1 - tt - neg) * inv;
        if (u0 > 20.0f)  u0 = 1.0f;
        if (u0 < -20.0f) u0 = 0.0f;
        if (u1 > 20.0f)  u1 = 1.0f;
        if (u1 < -20.0f) u1 = 0.0f;

        *(float2*)&outp[(size_t)(gy0 + ty) * WW + (gx0 + tx2)] = make_float2(u0, u1);
    }
}

extern "C" void kernel_launch(void* const* d_in, const int* in_sizes, int n_in,
                              void* d_out, int out_size, void* d_ws, size_t ws_size,
                              hipStream_t stream) {
    const float* x    = (const float*)d_in[0];
    const float* dx1  = (const float*)d_in[1];
    const float* dy1  = (const float*)d_in[2];
    const float* dx2  = (const float*)d_in[3];
    const float* dy2  = (const float*)d_in[4];
    const float* th   = (const float*)d_in[5];
    const float* amb  = (const float*)d_in[6];
    const int*   chix = (const int*)d_in[7];
    float* out = (float*)d_out;

    dim3 grid(WW / TS_X, HH / TS_Y, NN * MK);
    fern_bits_kernel<<<grid, 256, 0, stream>>>(x, dx1, dy1, dx2, dy2, th, amb, chix, out);
}